// PtrDecode_34677565948852
// MI455X (gfx1250) — compile-verified
//
#include <hip/hip_runtime.h>
#include <hip/hip_bf16.h>
#include <math.h>

typedef __attribute__((ext_vector_type(2))) float v2f;
typedef __attribute__((ext_vector_type(8))) float v8f;

#define HID 200
#define BB  16
#define TT  50
#define SS  200
#define VV  50000
#define OUTW 50200   // 50000 + 200

// ---------------- K0: Gi[b,t,j] = embed[y[bt]] . W_ih[j,:] + b_ih[j]  (j<600)
__global__ __launch_bounds__(256) void gi_kernel(const int* __restrict__ y,
        const float* __restrict__ embed, const float* __restrict__ W_ih,
        const float* __restrict__ b_ih, float* __restrict__ Gi) {
    int bt = blockIdx.x;                 // 0..799
    int tid = threadIdx.x;
    __shared__ float xrow[HID];
    int tok = y[bt];
    if (tid < HID) xrow[tid] = embed[(size_t)tok * HID + tid];
    __syncthreads();
    for (int j = tid; j < 3 * HID; j += 256) {
        const float* w = W_ih + (size_t)j * HID;
        float acc = b_ih[j];
        #pragma unroll 4
        for (int k = 0; k < HID; ++k) acc += xrow[k] * w[k];
        Gi[(size_t)bt * 600 + j] = acc;
    }
}

// ---------------- K1: GRU scan, one block per batch element
__global__ __launch_bounds__(640) void gru_kernel(const float* __restrict__ h1,
        const float* __restrict__ Gi, const float* __restrict__ W_hh,
        const float* __restrict__ b_hh, float* __restrict__ h2) {
    int b = blockIdx.x;
    int tid = threadIdx.x;
    __shared__ float h[HID];
    __shared__ float gh[3 * HID];
    if (tid < HID) h[tid] = h1[((size_t)b * (SS + 1) + SS) * HID + tid];  // h1[b,-1,:]
    __syncthreads();
    for (int t = 0; t < TT; ++t) {
        if (tid < 3 * HID) {
            const float* w = W_hh + (size_t)tid * HID;
            float acc = b_hh[tid];
            #pragma unroll 4
            for (int k = 0; k < HID; ++k) acc += h[k] * w[k];
            gh[tid] = acc;
        }
        __syncthreads();
        float hnew = 0.f;
        if (tid < HID) {
            const float* gi = Gi + ((size_t)b * TT + t) * 600;
            float r = 1.f / (1.f + expf(-(gi[tid]           + gh[tid])));
            float z = 1.f / (1.f + expf(-(gi[HID + tid]     + gh[HID + tid])));
            float n = tanhf(gi[2 * HID + tid] + r * gh[2 * HID + tid]);
            hnew = (1.f - z) * n + z * h[tid];
        }
        __syncthreads();
        if (tid < HID) {
            h[tid] = hnew;
            h2[((size_t)b * TT + t) * HID + tid] = hnew;
        }
        __syncthreads();
    }
}

// ---------------- K2: kv = h_enc @ W_attn + b ; q = h2 @ W_attn + b
__global__ __launch_bounds__(256) void proj_kernel(const float* __restrict__ h1,
        const float* __restrict__ h2, const float* __restrict__ W_attn,
        const float* __restrict__ b_attn, float* __restrict__ kv, float* __restrict__ q) {
    int row = blockIdx.x;                // 0..3999 : first 3200 kv rows, then 800 q rows
    int tid = threadIdx.x;
    __shared__ float in[HID];
    const float* src; float* dst;
    if (row < BB * SS) {
        int b = row / SS, s = row % SS;
        src = h1 + ((size_t)b * (SS + 1) + s) * HID;   // h_enc
        dst = kv + (size_t)row * HID;
    } else {
        int bt = row - BB * SS;
        src = h2 + (size_t)bt * HID;
        dst = q + (size_t)bt * HID;
    }
    if (tid < HID) in[tid] = src[tid];
    __syncthreads();
    if (tid < HID) {
        float acc = b_attn[tid];
        #pragma unroll 4
        for (int k = 0; k < HID; ++k) acc += in[k] * W_attn[k * HID + tid];
        dst[tid] = acc;
    }
}

// ---------------- K3: attention + gate + p1 output + s2 assembly
__global__ __launch_bounds__(256) void attn_kernel(const float* __restrict__ q,
        const float* __restrict__ kv, const float* __restrict__ h2,
        const float* __restrict__ W_gate, const float* __restrict__ b_gate,
        float* __restrict__ s2, float* __restrict__ gbuf, float* __restrict__ out) {
    int bt = blockIdx.x;                 // 0..799
    int b = bt / TT;
    int tid = threadIdx.x;
    __shared__ float qr[HID], sc[SS], aw[SS], cc[HID], red[256];
    if (tid < HID) qr[tid] = q[(size_t)bt * HID + tid];
    __syncthreads();
    const float scale = 7.0710678118654755e-2f;  // 1/sqrt(200)
    if (tid < SS) {
        const float* kr = kv + ((size_t)b * SS + tid) * HID;
        float acc = 0.f;
        #pragma unroll 4
        for (int k = 0; k < HID; ++k) acc += qr[k] * kr[k];
        sc[tid] = acc * scale;
    }
    __syncthreads();
    // max over S
    red[tid] = (tid < SS) ? sc[tid] : -INFINITY;
    __syncthreads();
    for (int s = 128; s > 0; s >>= 1) { if (tid < s) red[tid] = fmaxf(red[tid], red[tid + s]); __syncthreads(); }
    float m = red[0];
    __syncthreads();
    float e = (tid < SS) ? expf(sc[tid] - m) : 0.f;
    red[tid] = e;
    __syncthreads();
    for (int s = 128; s > 0; s >>= 1) { if (tid < s) red[tid] += red[tid + s]; __syncthreads(); }
    float Z = red[0];
    float lz = m + logf(Z);
    if (tid < SS) aw[tid] = e / Z;
    __syncthreads();
    // context c = a @ kv
    if (tid < HID) {
        const float* kb = kv + (size_t)b * SS * HID + tid;
        float acc = 0.f;
        #pragma unroll 4
        for (int s = 0; s < SS; ++s) acc += aw[s] * kb[(size_t)s * HID];
        cc[tid] = acc;
    }
    __syncthreads();
    // gate g = [h2,c] . W_gate + b_gate  (linear, no sigmoid per reference)
    float gp = 0.f;
    if (tid < HID) {
        float hv = h2[(size_t)bt * HID + tid];
        gp = hv * W_gate[tid] + cc[tid] * W_gate[HID + tid];
        s2[(size_t)bt * 400 + tid] = hv;
        s2[(size_t)bt * 400 + HID + tid] = cc[tid];
    }
    red[tid] = gp;
    __syncthreads();
    for (int s = 128; s > 0; s >>= 1) { if (tid < s) red[tid] += red[tid + s]; __syncthreads(); }
    float g = red[0] + b_gate[0];
    if (tid == 0) gbuf[bt] = g;
    if (tid < SS) {
        float p1 = sc[tid] - lz;
        out[(size_t)bt * OUTW + VV + tid] = (1.f - g) * p1;
    }
}

// ---------------- E1: logits = s2 @ W_out + b_out  via V_WMMA_F32_16X16X4_F32
// grid: (50 m-tiles, 196 n-groups of 256). Block = 8 waves, each wave does 16Mx32N.
__global__ __launch_bounds__(256) void logits_kernel(const float* __restrict__ s2,
        const float* __restrict__ W_out, const float* __restrict__ b_out,
        float* __restrict__ out) {
    __shared__ float sA[16][402];        // 402 stride: bank-spread + 8B aligned pairs
    const int m0 = blockIdx.x * 16;
    const int n_base = blockIdx.y * 256;
    const int tid = threadIdx.x;
    const int lane = tid & 31;
    const int wave = tid >> 5;
    // stage A tile (16 x 400) into LDS
    for (int idx = tid; idx < 16 * 400; idx += 256) {
        int r = idx / 400, k = idx - r * 400;
        sA[r][k] = s2[(size_t)(m0 + r) * 400 + k];
    }
    __syncthreads();

    const int mrow = lane & 15;          // A row / B+D column index
    const int hi   = lane >> 4;
    const int khi  = hi * 2;             // K sub-rows 0,1 (lo half) / 2,3 (hi half)
    const int n0   = n_base + wave * 32;
    const int colA = n0 + mrow;
    const int colB = n0 + 16 + mrow;
    const int ldA  = colA < (VV - 1) ? colA : (VV - 1);   // clamp OOB loads
    const int ldB  = colB < (VV - 1) ? colB : (VV - 1);

    v8f c0 = {}; v8f c1 = {};
    #pragma unroll 4
    for (int kb = 0; kb < 400; kb += 4) {
        v2f a;
        a.x = sA[mrow][kb + khi];
        a.y = sA[mrow][kb + khi + 1];
        const float* wp = W_out + (size_t)(kb + khi) * VV;
        v2f bA, bB;
        bA.x = wp[ldA]; bA.y = wp[VV + ldA];
        bB.x = wp[ldB]; bB.y = wp[VV + ldB];
        c0 = __builtin_amdgcn_wmma_f32_16x16x4_f32(false, a, false, bA, (short)0, c0, false, false);
        c1 = __builtin_amdgcn_wmma_f32_16x16x4_f32(false, a, false, bB, (short)0, c1, false, false);
    }

    const int rbase = hi * 8;            // D rows: r + 8*hi
    if (colA < VV) {
        float bias = b_out[colA];
        #pragma unroll
        for (int r = 0; r < 8; ++r)
            out[(size_t)(m0 + rbase + r) * OUTW + colA] = c0[r] + bias;
    }
    if (colB < VV) {
        float bias = b_out[colB];
        #pragma unroll
        for (int r = 0; r < 8; ++r)
            out[(size_t)(m0 + rbase + r) * OUTW + colB] = c1[r] + bias;
    }
}

// ---------------- E2: in-place log-softmax over V and scale by g
__global__ __launch_bounds__(256) void finish_kernel(float* __restrict__ out,
        const float* __restrict__ gbuf) {
    int row = blockIdx.x;                // 0..799
    int tid = threadIdx.x;
    float g = gbuf[row];
    float* p = out + (size_t)row * OUTW;
    __shared__ float red[256];
    float m = -INFINITY;
    for (int j = tid; j < VV; j += 256) m = fmaxf(m, p[j]);
    red[tid] = m;
    __syncthreads();
    for (int s = 128; s > 0; s >>= 1) { if (tid < s) red[tid] = fmaxf(red[tid], red[tid + s]); __syncthreads(); }
    m = red[0];
    __syncthreads();
    float sum = 0.f;
    for (int j = tid; j < VV; j += 256) sum += expf(p[j] - m);
    red[tid] = sum;
    __syncthreads();
    for (int s = 128; s > 0; s >>= 1) { if (tid < s) red[tid] += red[tid + s]; __syncthreads(); }
    float lz = m + logf(red[0]);
    __syncthreads();
    for (int j = tid; j < VV; j += 256) p[j] = g * (p[j] - lz);
}

extern "C" void kernel_launch(void* const* d_in, const int* in_sizes, int n_in,
                              void* d_out, int out_size, void* d_ws, size_t ws_size,
                              hipStream_t stream) {
    const int*   y      = (const int*)  d_in[0];
    const float* h1     = (const float*)d_in[1];
    const float* embed  = (const float*)d_in[2];
    const float* W_ih   = (const float*)d_in[3];
    const float* W_hh   = (const float*)d_in[4];
    const float* b_ih   = (const float*)d_in[5];
    const float* b_hh   = (const float*)d_in[6];
    const float* W_attn = (const float*)d_in[7];
    const float* b_attn = (const float*)d_in[8];
    const float* W_gate = (const float*)d_in[9];
    const float* b_gate = (const float*)d_in[10];
    const float* W_out  = (const float*)d_in[11];
    const float* b_out  = (const float*)d_in[12];
    float* out = (float*)d_out;
    float* ws  = (float*)d_ws;

    float* Gi = ws;                  // 800*600   = 480000
    float* h2 = Gi + 480000;         // 800*200   = 160000
    float* kv = h2 + 160000;         // 3200*200  = 640000
    float* q  = kv + 640000;         // 800*200   = 160000
    float* s2 = q  + 160000;         // 800*400   = 320000
    float* gb = s2 + 320000;         // 800
    // total ~7.05 MB of workspace

    gi_kernel    <<<800, 256, 0, stream>>>(y, embed, W_ih, b_ih, Gi);
    gru_kernel   <<<16, 640, 0, stream>>>(h1, Gi, W_hh, b_hh, h2);
    proj_kernel  <<<4000, 256, 0, stream>>>(h1, h2, W_attn, b_attn, kv, q);
    attn_kernel  <<<800, 256, 0, stream>>>(q, kv, h2, W_gate, b_gate, s2, gb, out);
    logits_kernel<<<dim3(50, 196), 256, 0, stream>>>(s2, W_out, b_out, out);
    finish_kernel<<<800, 256, 0, stream>>>(out, gb);
}